// Torch_RNN_15547781611650
// MI455X (gfx1250) — compile-verified
//
#include <hip/hip_runtime.h>
#include <hip/hip_bf16.h>

typedef float v2f __attribute__((ext_vector_type(2)));
typedef float v8f __attribute__((ext_vector_type(8)));
typedef unsigned int v4u __attribute__((ext_vector_type(4)));
typedef int v4i __attribute__((ext_vector_type(4)));
typedef int v8i __attribute__((ext_vector_type(8)));

#if __has_builtin(__builtin_amdgcn_tensor_load_to_lds) && \
    __has_builtin(__builtin_amdgcn_s_wait_tensorcnt)
#define USE_TDM 1
#else
#define USE_TDM 0
#endif

namespace {

constexpr int T_LEN = 512;
constexpr int D = 30;
constexpr int H1 = 15;
constexpr int S32 = 34;  // padded row stride (floats) of 32-col tiles: bank-conflict-free
constexpr int S16 = 18;  // padded row stride (floats) of 16-col tiles

__device__ inline float fast_tanh(float v) {
  // tanh(v) = 1 - 2/(exp(2v)+1);  exp(2v) = exp2(v * 2*log2(e))
  float e = __builtin_amdgcn_exp2f(v * 2.885390081777927f);
  return 1.0f - 2.0f * __builtin_amdgcn_rcpf(e + 1.0f);
}

__device__ inline void lds_fence() {
  asm volatile("s_wait_dscnt 0" ::: "memory");
}

__device__ inline v8f zero8() { v8f z = {}; return z; }

__device__ inline v8f wmma4(v2f a, v2f b, v8f c) {
  // D = A(16x4 f32) * B(4x16 f32) + C(16x16 f32)
  return __builtin_amdgcn_wmma_f32_16x16x4_f32(false, a, false, b, (short)0, c,
                                               false, false);
}

#if USE_TDM
typedef __attribute__((address_space(3))) float lds_float;

__device__ inline unsigned lds_off(const float* p) {
  return (unsigned)(uintptr_t)(lds_float*)p;
}

// One TDM descriptor moves a 16-row x 32-col f32 tile: rows are batch rows of
// x (stride T*D elements apart), columns 30..31 are OOB vs tensor_dim0=30 and
// zero-fill; LDS pad of 2 DWORDs per 32-DWORD row yields the stride-34 layout.
__device__ inline void tdm_load_x_tile(const float* gsrc, unsigned ldsaddr) {
  const unsigned long long ga = (unsigned long long)(uintptr_t)gsrc;
  v4u g0;
  g0[0] = 1u;                                 // count=1 (valid), user mode
  g0[1] = ldsaddr;                            // lds_addr (bytes)
  g0[2] = (unsigned)ga;                       // global_addr[31:0]
  g0[3] = (unsigned)(ga >> 32) | (2u << 30);  // global_addr[56:32] | type=2
  v8i g1;
  g1[0] = (2 << 16)       // data_size = 4 bytes
        | (1 << 20)       // pad_enable
        | (4 << 22)       // pad_interval: every 32 DWORDs (one row)
        | (1 << 25);      // pad_amount: 2 DWORDs  -> LDS row stride 34 floats
  g1[1] = (int)(30u << 16);  // tensor_dim0 = 30 (bits 79:48)
  g1[2] = (int)(16u << 16);  // tensor_dim1 = 16 (bits 111:80)
  g1[3] = (int)(32u << 16);  // tile_dim0  = 32 (bits 127:112)
  g1[4] = 16;                // tile_dim1  = 16 (bits 143:128)
  g1[5] = T_LEN * D;         // tensor_dim0_stride (bits 207:160): batch rows
  g1[6] = 0;
  g1[7] = 0;
  const v4i z4 = {};
  const v8i z8 = {};
  // 6-arg form (this toolchain): (g0, g1, g2, g3, extra, cpol)
  __builtin_amdgcn_tensor_load_to_lds(g0, g1, z4, z4, z8, 0);
}

__device__ inline void tdm_wait_all_but_one() {
  __builtin_amdgcn_s_wait_tensorcnt(1);
  asm volatile("" ::: "memory");  // keep LDS reads below the wait
}
#endif

} // namespace

// One wave (32 lanes) owns a 16-batch tile and walks the full T=512 recurrence.
// Software-pipelined: iteration t computes L2(t)+head(t) AND L1(t+1) in one
// straight-line block (8 independent WMMA chains of depth 8).
// LDS tiles (per wave), all bank-conflict-free padded strides:
//   XT[2]  : x_{t}/x_{t+1} staging, TDM double buffer (16 x 32, stride 34)
//   H1R    : h1 state as L1 recurrent input
//   H1X    : h1 state as L2 input
//   H2T    : h2 state (L2 recurrent input + head input)
//   A1T/A2T: head activations (16 x 16, stride 18)
__global__ __launch_bounds__(32) void rnn2_mlp_fused(
    const float* __restrict__ x,
    const float* __restrict__ Wih1, const float* __restrict__ Whh1,
    const float* __restrict__ bih1, const float* __restrict__ bhh1,
    const float* __restrict__ Wih2, const float* __restrict__ Whh2,
    const float* __restrict__ bih2, const float* __restrict__ bhh2,
    const float* __restrict__ W1, const float* __restrict__ b1,
    const float* __restrict__ W2, const float* __restrict__ b2,
    const float* __restrict__ W3, const float* __restrict__ b3,
    float* __restrict__ out)
{
  __shared__ __align__(16) float XT[2][16 * S32];
  __shared__ __align__(16) float H1R[16 * S32];
  __shared__ __align__(16) float H1X[16 * S32];
  __shared__ __align__(16) float H2T[16 * S32];
  __shared__ __align__(16) float A1T[16 * S16];
  __shared__ __align__(16) float A2T[16 * S16];

  const int lane  = threadIdx.x;      // 0..31
  const int nlo   = lane & 15;        // N (or M) index within 16
  const int khalf = (lane >> 4) & 1;  // lane-half selects K pair / M upper half
  const int base  = blockIdx.x * 16;  // first batch row of this tile

  // ---- zero LDS (pad columns must start as 0; h-state starts as 0) ----
  for (int i = lane; i < 2 * 16 * S32; i += 32) (&XT[0][0])[i] = 0.f;
  for (int i = lane; i < 16 * S32; i += 32) {
    H1R[i] = 0.f; H1X[i] = 0.f; H2T[i] = 0.f;
  }
  for (int i = lane; i < 16 * S16; i += 32) { A1T[i] = 0.f; A2T[i] = 0.f; }

  // ---- preload all weight B-fragments into VGPRs (resident whole kernel) ----
  // B fragment (4x16 f32): lane<16 -> K=4c+{0,1}, lane>=16 -> K=4c+{2,3},
  // N = lane%16 (+16 per n-tile); zeros in padding rows/cols.
  v2f B1[16][2], B2[16][2], BH1[8], BH2[4];
#pragma unroll
  for (int c = 0; c < 16; ++c) {
#pragma unroll
    for (int nt = 0; nt < 2; ++nt) {
      const int n = nlo + 16 * nt;
      v2f f1 = {}, f2 = {};
#pragma unroll
      for (int r = 0; r < 2; ++r) {
        const int k = 4 * c + 2 * khalf + r;
        float w1v = 0.f, w2v = 0.f;
        if (n < D) {
          if (k < D) {                        // input-projection half: Wih^T
            w1v = Wih1[n * D + k];
            w2v = Wih2[n * D + k];
          } else if (k >= 32 && k < 32 + D) { // recurrent half: Whh^T
            w1v = Whh1[n * D + (k - 32)];
            w2v = Whh2[n * D + (k - 32)];
          }
        }
        f1[r] = w1v;
        f2[r] = w2v;
      }
      B1[c][nt] = f1;
      B2[c][nt] = f2;
    }
  }
#pragma unroll
  for (int c = 0; c < 8; ++c) {   // head layer 1: K=32 (h2 padded), N=16
    v2f f = {};
#pragma unroll
    for (int r = 0; r < 2; ++r) {
      const int k = 4 * c + 2 * khalf + r;
      f[r] = (nlo < H1 && k < D) ? W1[nlo * D + k] : 0.f;
    }
    BH1[c] = f;
  }
#pragma unroll
  for (int c = 0; c < 4; ++c) {   // head layer 2: K=16 (a1 padded), N=16
    v2f f = {};
#pragma unroll
    for (int r = 0; r < 2; ++r) {
      const int k = 4 * c + 2 * khalf + r;
      f[r] = (nlo < H1 && k < H1) ? W2[nlo * H1 + k] : 0.f;
    }
    BH2[c] = f;
  }

  // ---- per-lane bias values (by output column = lane%16 [+16]) ----
  float bias1[2], bias2[2];
#pragma unroll
  for (int nt = 0; nt < 2; ++nt) {
    const int n = nlo + 16 * nt;
    bias1[nt] = (n < D) ? (bih1[n] + bhh1[n]) : 0.f;
    bias2[nt] = (n < D) ? (bih2[n] + bhh2[n]) : 0.f;
  }
  const float biasH1 = (nlo < H1) ? b1[nlo] : 0.f;
  const float biasH2 = (nlo < H1) ? b2[nlo] : 0.f;
  float w3r[16];
#pragma unroll
  for (int j = 0; j < 16; ++j) w3r[j] = (j < H1) ? W3[j] : 0.f;
  const float b3v = b3[0];

  // ---- helpers ----
  auto afrag = [&](const float* t0, int stride, int c) -> v2f {
    // A fragment (16x4 f32): lane<16 holds K=4c,4c+1; lane>=16 holds K=4c+2,4c+3
    return *(const v2f*)(t0 + nlo * stride + 4 * c + 2 * khalf);
  };
  auto store_one = [&](v8f acc, float bias, float* dst, int stride, int colbase) {
    const int row0 = khalf * 8;
    const int col = colbase + nlo;
#pragma unroll
    for (int r = 0; r < 8; ++r)
      dst[(row0 + r) * stride + col] = fast_tanh(acc[r] + bias);
  };
  auto store_dual = [&](v8f acc, float bias, float* d0, float* d1, int colbase) {
    const int row0 = khalf * 8;
    const int col = colbase + nlo;
#pragma unroll
    for (int r = 0; r < 8; ++r) {
      const float v = fast_tanh(acc[r] + bias);
      d0[(row0 + r) * S32 + col] = v;
      d1[(row0 + r) * S32 + col] = v;
    }
  };

  const float* xb = x + (size_t)base * T_LEN * D;

#if !USE_TDM
  // Fallback staging: per-lane constant offsets, scalar base walk.
  int goff[15], loff[15];
#pragma unroll
  for (int j = 0; j < 15; ++j) {
    const int i = lane + 32 * j;
    const int m = i / D;
    const int d = i - D * m;
    goff[j] = m * (T_LEN * D) + d;
    loff[j] = m * S32 + d;
  }
  float xr[15];
  auto xload = [&](int t) {
    const float* xt = xb + (size_t)t * D;
#pragma unroll
    for (int j = 0; j < 15; ++j) xr[j] = xt[goff[j]];
  };
  auto xstore = [&](float* dst) {
#pragma unroll
    for (int j = 0; j < 15; ++j) dst[loff[j]] = xr[j];
  };
#endif

  float* const orow = out + (size_t)(base + nlo) * T_LEN;

  // ================= prologue: stage x_0/x_1, run L1(0) =====================
  lds_fence();
#if USE_TDM
  tdm_load_x_tile(xb, lds_off(&XT[0][0]));
  tdm_load_x_tile(xb + D, lds_off(&XT[1][0]));
  tdm_wait_all_but_one();              // x_0 resident
#else
  xload(0); xstore(&XT[0][0]);
  xload(1); xstore(&XT[1][0]);
  lds_fence();
#endif
  {
    v8f a0e = zero8(), a0o = zero8(), a1e = zero8(), a1o = zero8();
#pragma unroll
    for (int c = 0; c < 8; ++c) {
      const v2f ax = afrag(&XT[0][0], S32, c);  // x_0
      const v2f ah = afrag(H1R, S32, c);        // h1_{-1} = 0
      if (c & 1) {
        a0o = wmma4(ax, B1[c][0], a0o);     a1o = wmma4(ax, B1[c][1], a1o);
        a0o = wmma4(ah, B1[c + 8][0], a0o); a1o = wmma4(ah, B1[c + 8][1], a1o);
      } else {
        a0e = wmma4(ax, B1[c][0], a0e);     a1e = wmma4(ax, B1[c][1], a1e);
        a0e = wmma4(ah, B1[c + 8][0], a0e); a1e = wmma4(ah, B1[c + 8][1], a1e);
      }
    }
    store_dual(a0e + a0o, bias1[0], H1R, H1X, 0);   // h1_0
    store_dual(a1e + a1o, bias1[1], H1R, H1X, 16);
    lds_fence();
  }

  // ================= serial time loop ======================================
  for (int t = 0; t < T_LEN; ++t) {
    // stage x_{t+2} into the buffer whose content (x_t) is now dead
    const int tn = (t + 2 < T_LEN) ? t + 2 : T_LEN - 1;
    float* xnew = &XT[t & 1][0];
    const float* xcur = &XT[(t + 1) & 1][0];  // x_{t+1} for L1(t+1)
#if USE_TDM
    tdm_load_x_tile(xb + (size_t)tn * D, lds_off(xnew));
    tdm_wait_all_but_one();   // previous TDM (x_{t+1}) complete; new one flies
#else
    xload(tn);
#endif

    // ---- fused block: L2(t) over {H1X,H2T}  +  L1(t+1) over {x_{t+1},H1R} --
    // 8 independent WMMA chains of depth 8 -> scheduler interleaves.
    v8f l2_0e = zero8(), l2_0o = zero8(), l2_1e = zero8(), l2_1o = zero8();
    v8f l1_0e = zero8(), l1_0o = zero8(), l1_1e = zero8(), l1_1o = zero8();
#pragma unroll
    for (int c = 0; c < 8; ++c) {
      const v2f a2x = afrag(H1X, S32, c);   // L2 K-chunks 0..7  (h1_t)
      const v2f a2h = afrag(H2T, S32, c);   // L2 K-chunks 8..15 (h2_{t-1})
      const v2f a1x = afrag(xcur, S32, c);  // L1 K-chunks 0..7  (x_{t+1})
      const v2f a1h = afrag(H1R, S32, c);   // L1 K-chunks 8..15 (h1_t)
      if (c & 1) {
        l2_0o = wmma4(a2x, B2[c][0], l2_0o);     l2_1o = wmma4(a2x, B2[c][1], l2_1o);
        l2_0o = wmma4(a2h, B2[c + 8][0], l2_0o); l2_1o = wmma4(a2h, B2[c + 8][1], l2_1o);
        l1_0o = wmma4(a1x, B1[c][0], l1_0o);     l1_1o = wmma4(a1x, B1[c][1], l1_1o);
        l1_0o = wmma4(a1h, B1[c + 8][0], l1_0o); l1_1o = wmma4(a1h, B1[c + 8][1], l1_1o);
      } else {
        l2_0e = wmma4(a2x, B2[c][0], l2_0e);     l2_1e = wmma4(a2x, B2[c][1], l2_1e);
        l2_0e = wmma4(a2h, B2[c + 8][0], l2_0e); l2_1e = wmma4(a2h, B2[c + 8][1], l2_1e);
        l1_0e = wmma4(a1x, B1[c][0], l1_0e);     l1_1e = wmma4(a1x, B1[c][1], l1_1e);
        l1_0e = wmma4(a1h, B1[c + 8][0], l1_0e); l1_1e = wmma4(a1h, B1[c + 8][1], l1_1e);
      }
    }
    // h2_t -> H2T (head input now; L2(t+1) recurrent input)
    store_one(l2_0e + l2_0o, bias2[0], H2T, S32, 0);
    store_one(l2_1e + l2_1o, bias2[1], H2T, S32, 16);
    // h1_{t+1} -> H1R (L1(t+2)) and H1X (L2(t+1))
    store_dual(l1_0e + l1_0o, bias1[0], H1R, H1X, 0);
    store_dual(l1_1e + l1_1o, bias1[1], H1R, H1X, 16);
    lds_fence();

    // ---- head layer 1: tanh(h2_t * W1^T + b1) ----
    {
      v8f he = zero8(), ho = zero8();
#pragma unroll
      for (int c = 0; c < 8; ++c) {
        const v2f a = afrag(H2T, S32, c);
        if (c & 1) ho = wmma4(a, BH1[c], ho);
        else       he = wmma4(a, BH1[c], he);
      }
      store_one(he + ho, biasH1, A1T, S16, 0);
    }
    lds_fence();

    // ---- head layer 2: tanh(a1 * W2^T + b2) ----
    {
      v8f he = zero8(), ho = zero8();
#pragma unroll
      for (int c = 0; c < 4; ++c) {
        const v2f a = afrag(A1T, S16, c);
        if (c & 1) ho = wmma4(a, BH2[c], ho);
        else       he = wmma4(a, BH2[c], he);
      }
      store_one(he + ho, biasH2, A2T, S16, 0);
    }
    lds_fence();

    // ---- head layer 3: 15-wide dot + bias; lane m handles batch row m ----
    if (lane < 16) {
      const float* row = &A2T[lane * S16];
      float y = b3v;
#pragma unroll
      for (int j = 0; j < 16; ++j) y += row[j] * w3r[j];
      orow[t] = y;
    }
    lds_fence();

#if !USE_TDM
    xstore(xnew);  // stage x_{t+2}
    lds_fence();
#endif
  }
}

extern "C" void kernel_launch(void* const* d_in, const int* in_sizes, int n_in,
                              void* d_out, int out_size, void* d_ws, size_t ws_size,
                              hipStream_t stream) {
  (void)in_sizes; (void)n_in; (void)out_size; (void)d_ws; (void)ws_size;
  const float* x    = (const float*)d_in[0];
  const float* Wih1 = (const float*)d_in[1];
  const float* Whh1 = (const float*)d_in[2];
  const float* bih1 = (const float*)d_in[3];
  const float* bhh1 = (const float*)d_in[4];
  const float* Wih2 = (const float*)d_in[5];
  const float* Whh2 = (const float*)d_in[6];
  const float* bih2 = (const float*)d_in[7];
  const float* bhh2 = (const float*)d_in[8];
  const float* W1   = (const float*)d_in[9];
  const float* b1   = (const float*)d_in[10];
  const float* W2   = (const float*)d_in[11];
  const float* b2   = (const float*)d_in[12];
  const float* W3   = (const float*)d_in[13];
  const float* b3   = (const float*)d_in[14];

  dim3 grid(4096 / 16);  // 256 independent 16-batch tiles, one wave each
  dim3 block(32);
  rnn2_mlp_fused<<<grid, block, 0, stream>>>(x, Wih1, Whh1, bih1, bhh1,
                                             Wih2, Whh2, bih2, bhh2,
                                             W1, b1, W2, b2, W3, b3,
                                             (float*)d_out);
}